// Stgc_block_71502615544271
// MI455X (gfx1250) — compile-verified
//
#include <hip/hip_runtime.h>
#include <hip/hip_bf16.h>

// ---------------------------------------------------------------------------
// ST-GCN block for MI455X (gfx1250, wave32), fp32 WMMA 16x16x4 path.
//   N=16, C=256, T=300, V=25, K=3 spatial kernels, TK=9 temporal taps.
// Memory-bound (~615 MB @ 23.3 TB/s); fp32 WMMA keeps reference precision.
// Kernel 2 uses GLOBAL_LOAD_ASYNC_TO_LDS_B128 + ASYNCcnt double buffering.
// ---------------------------------------------------------------------------

#define N_    16
#define CIN   256
#define COUT  256
#define T_    300
#define V_    25
#define KS    3
#define TKS   9
#define COLS  (T_ * V_)        // 7500 flattened (t,v) columns per (n, channel)
#define BN_EPS 1e-5f

typedef __attribute__((ext_vector_type(2))) float v2f;
typedef __attribute__((ext_vector_type(8))) float v8f;

__device__ __forceinline__ v8f wmma_f32_16x16x4(v2f a, v2f b, v8f c) {
    // D = A(16x4) * B(4x16) + C(16x16), all fp32, wave32.
    return __builtin_amdgcn_wmma_f32_16x16x4_f32(
        /*neg_a=*/false, a, /*neg_b=*/false, b,
        /*c_mod=*/(short)0, c, /*reuse_a=*/false, /*reuse_b=*/false);
}

// ===========================================================================
// Kernel 1: fused  1x1-conv -> spatial graph contraction -> BN1 -> ReLU
//   One block per (n, t).  8 waves; wave w owns output channels [32w, 32w+32).
//   GEMM1: Y_k(32x32) = W_k(32x256) * X(256x32padded)   (WMMA, K=256)
//   GEMM2: S        += Y_k(32x32)  * AM_k(32x32padded)  (WMMA, K=32)
// ===========================================================================
__global__ __launch_bounds__(256)
void sgc_bn_relu_kernel(const float* __restrict__ x,
                        const float* __restrict__ A,
                        const float* __restrict__ M,
                        const float* __restrict__ conv_w,   // (K*COUT, CIN)
                        const float* __restrict__ conv_b,   // (K*COUT)
                        const float* __restrict__ g1, const float* __restrict__ b1,
                        const float* __restrict__ m1, const float* __restrict__ v1,
                        float* __restrict__ h1)             // (N,COUT,T,V)
{
    const int n = blockIdx.y;
    const int t = blockIdx.x;
    const int tid  = threadIdx.x;
    const int wave = tid >> 5;
    const int lane = tid & 31;
    const int lm   = lane & 15;        // M/N index within 16
    const int lhi  = lane >> 4;        // 0: K=0,1 / rows 0..7 ; 1: K=2,3 / rows 8..15

    __shared__ float Xs[CIN][32];      // X = x[n,:,t,:], cols padded 25->32 (zeros)
    __shared__ float AMs[KS][32][32];  // A*M, zero padded
    __shared__ float Ssum[KS][32];     // column sums of AM (for folded conv bias)
    __shared__ float Ys[8][32][32];    // per-wave Y tile scratch (C-layout -> A-layout)

    // ---- stage X: thread tid loads channel row tid (25 contiguous floats) ----
    {
        const float* xr = x + (((size_t)n * CIN + tid) * T_ + t) * V_;
        #pragma unroll
        for (int v = 0; v < V_; ++v) Xs[tid][v] = xr[v];
        #pragma unroll
        for (int v = V_; v < 32; ++v) Xs[tid][v] = 0.0f;
    }
    // ---- stage AM = A * M, zero padded to 32x32 per k ----
    for (int i = tid; i < KS * 32 * 32; i += 256) {
        const int k = i >> 10, r = (i >> 5) & 31, c = i & 31;
        float val = 0.0f;
        if (r < V_ && c < V_) {
            const int idx = (k * V_ + r) * V_ + c;
            val = A[idx] * M[idx];
        }
        AMs[k][r][c] = val;
    }
    __syncthreads();
    // ---- column sums of AM (bias folding: sum_v AM[k][v][w]) ----
    if (tid < KS * 32) {
        const int k = tid >> 5, c = tid & 31;
        float s = 0.0f;
        #pragma unroll
        for (int v = 0; v < V_; ++v) s += AMs[k][v][c];
        Ssum[k][c] = s;
    }
    __syncthreads();

    v8f sac[2][2] = {};   // sgc accumulators: 2 row-tiles x 2 col-tiles

    for (int k = 0; k < KS; ++k) {
        // ---------------- GEMM1: Y_k = W_k * X ----------------
        v8f yac[2][2] = {};
        const float* Wk = conv_w + (size_t)(k * COUT + wave * 32) * CIN;
        for (int kk = 0; kk < CIN / 4; ++kk) {
            const int kb = kk * 4 + lhi * 2;   // this lane's K pair
            v2f bf[2];
            #pragma unroll
            for (int ct = 0; ct < 2; ++ct) {
                bf[ct][0] = Xs[kb    ][ct * 16 + lm];
                bf[ct][1] = Xs[kb + 1][ct * 16 + lm];
            }
            v2f af[2];
            #pragma unroll
            for (int rt = 0; rt < 2; ++rt)     // contiguous K-pair -> 64-bit load
                af[rt] = *(const v2f*)(Wk + (size_t)(rt * 16 + lm) * CIN + kb);
            #pragma unroll
            for (int rt = 0; rt < 2; ++rt)
                #pragma unroll
                for (int ct = 0; ct < 2; ++ct)
                    yac[rt][ct] = wmma_f32_16x16x4(af[rt], bf[ct], yac[rt][ct]);
        }
        // ---- C-layout -> LDS (wave-private tile, no cross-wave hazard) ----
        #pragma unroll
        for (int rt = 0; rt < 2; ++rt)
            #pragma unroll
            for (int ct = 0; ct < 2; ++ct)
                #pragma unroll
                for (int i = 0; i < 8; ++i)
                    Ys[wave][rt * 16 + lhi * 8 + i][ct * 16 + lm] = yac[rt][ct][i];
        asm volatile("s_wait_dscnt 0" ::: "memory");   // LDS RAW inside the wave

        // ---------------- GEMM2: S += Y_k * AM_k  (K = 32, zero padded) -------
        for (int kk2 = 0; kk2 < 8; ++kk2) {
            const int kb2 = kk2 * 4 + lhi * 2;
            v2f a2[2];
            #pragma unroll
            for (int rt = 0; rt < 2; ++rt)
                a2[rt] = *(const v2f*)(&Ys[wave][rt * 16 + lm][kb2]);
            v2f b2[2];
            #pragma unroll
            for (int ct = 0; ct < 2; ++ct) {
                b2[ct][0] = AMs[k][kb2    ][ct * 16 + lm];
                b2[ct][1] = AMs[k][kb2 + 1][ct * 16 + lm];
            }
            #pragma unroll
            for (int rt = 0; rt < 2; ++rt)
                #pragma unroll
                for (int ct = 0; ct < 2; ++ct)
                    sac[rt][ct] = wmma_f32_16x16x4(a2[rt], b2[ct], sac[rt][ct]);
        }
    }

    // ---------------- epilogue: + folded conv bias, BN1, ReLU, store ----------
    #pragma unroll
    for (int rt = 0; rt < 2; ++rt) {
        #pragma unroll
        for (int ct = 0; ct < 2; ++ct) {
            #pragma unroll
            for (int i = 0; i < 8; ++i) {
                const int c = wave * 32 + rt * 16 + lhi * 8 + i;
                const int w = ct * 16 + lm;
                if (w < V_) {
                    float val = sac[rt][ct][i];
                    #pragma unroll
                    for (int k = 0; k < KS; ++k)
                        val += conv_b[k * COUT + c] * Ssum[k][w];
                    const float sc = g1[c] * rsqrtf(v1[c] + BN_EPS);
                    val = fmaxf(val * sc + (b1[c] - m1[c] * sc), 0.0f);
                    h1[(((size_t)n * COUT + c) * T_ + t) * V_ + w] = val;
                }
            }
        }
    }
}

// ===========================================================================
// Kernel 2: temporal conv (9 taps) as 9 column-shifted GEMMs + BN2 + ReLU + x.
//   Block = (n, 64-column tile of flattened (t,v)).  Shift by dt is a shift of
//   25*dt flattened columns (v untouched).  8 waves; wave owns 32 out-channels
//   (2 row tiles) x 4 col tiles of persistent fp32 accumulators.
//   h1 slabs staged with GLOBAL_LOAD_ASYNC_TO_LDS_B128, double buffered on
//   ASYNCcnt so slab(cc+1) streams in while slab(cc) feeds the WMMAs.
// ===========================================================================
#define CTILE 64
#define HALO  100                 // 4 taps * 25 columns each side
#define LDSW  (CTILE + 2 * HALO) // 264 valid columns (multiple of 4)
#define LDSP  272                 // padded LDS row stride (16B aligned)
#define NG    (LDSW / 4)          // 66 float4 groups per row

__global__ __launch_bounds__(256)
void tconv_bn_relu_res_kernel(const float* __restrict__ h1,
                              const float* __restrict__ tw,  // (COUT, COUT, 9, 1)
                              const float* __restrict__ tb,
                              const float* __restrict__ g2, const float* __restrict__ b2,
                              const float* __restrict__ m2, const float* __restrict__ v2,
                              const float* __restrict__ x,
                              float* __restrict__ out)
{
    const int n    = blockIdx.y;
    const int col0 = blockIdx.x * CTILE;
    const int tid  = threadIdx.x;
    const int wave = tid >> 5;
    const int lane = tid & 31;
    const int lm   = lane & 15;
    const int lhi  = lane >> 4;
    const int orow0 = wave * 32;

    __shared__ float Hs[2][16][LDSP];   // double-buffered 16 x 264 slabs

    // Issue one chunk's slab as async B128 loads straight into LDS buffer b.
    // Geometry guarantees: global col = col0-100+4*gi is 0 mod 4 and each
    // 4-float group is entirely in [0,7500) or entirely outside -> exact
    // zero padding via the fixup pass, 16B-aligned global & LDS addresses.
    auto stage_async = [&](int cc2, int b) {
        const int cibase2 = cc2 * 16;
        for (int g = tid; g < 16 * NG; g += 256) {
            const int r  = g / NG;
            const int cI = (g - r * NG) * 4;
            long gc = (long)col0 - HALO + cI;
            if (gc < 0 || gc >= COLS) gc = 0;        // clamped; zeroed in fixup
            const float* gp = h1 + ((size_t)n * COUT + cibase2 + r) * COLS + gc;
            const unsigned lo = (unsigned)(uintptr_t)(&Hs[b][r][cI]);
            asm volatile("global_load_async_to_lds_b128 %0, %1, off"
                         :: "v"(lo), "v"(gp) : "memory");
        }
    };
    // Overwrite out-of-range groups with zeros (after s_wait_asynccnt 0).
    auto fixup_zeros = [&](int b) {
        for (int g = tid; g < 16 * NG; g += 256) {
            const int r  = g / NG;
            const int cI = (g - r * NG) * 4;
            const long gc = (long)col0 - HALO + cI;
            if (gc < 0 || gc >= COLS) {
                Hs[b][r][cI + 0] = 0.0f; Hs[b][r][cI + 1] = 0.0f;
                Hs[b][r][cI + 2] = 0.0f; Hs[b][r][cI + 3] = 0.0f;
            }
        }
    };

    v8f acc[2][4] = {};

    stage_async(0, 0);                                   // prologue: chunk 0

    for (int cc = 0; cc < COUT / 16; ++cc) {
        const int cur = cc & 1;
        const int cibase = cc * 16;

        asm volatile("s_wait_asynccnt 0x0" ::: "memory"); // our slab landed
        fixup_zeros(cur);                                 // exact T-edge zeros
        __syncthreads();                                  // all waves: data ready,
                                                          // other buffer consumed
        if (cc + 1 < COUT / 16)
            stage_async(cc + 1, cur ^ 1);                 // overlap with compute

        // ---- 9 taps x 4 K-steps of fp32 WMMA on Hs[cur] ----
        for (int dt = 0; dt < TKS; ++dt) {
            #pragma unroll
            for (int kk = 0; kk < 4; ++kk) {
                const int kb = kk * 4 + lhi * 2;
                v2f af[2];
                #pragma unroll
                for (int rt = 0; rt < 2; ++rt) {
                    const int o = orow0 + rt * 16 + lm;
                    const float* wp = tw + ((size_t)o * COUT + cibase + kb) * TKS + dt;
                    af[rt][0] = wp[0];
                    af[rt][1] = wp[TKS];
                }
                v2f bf[4];
                #pragma unroll
                for (int ct = 0; ct < 4; ++ct) {
                    const int cI = ct * 16 + lm + 25 * dt;
                    bf[ct][0] = Hs[cur][kb    ][cI];
                    bf[ct][1] = Hs[cur][kb + 1][cI];
                }
                #pragma unroll
                for (int rt = 0; rt < 2; ++rt)
                    #pragma unroll
                    for (int ct = 0; ct < 4; ++ct)
                        acc[rt][ct] = wmma_f32_16x16x4(af[rt], bf[ct], acc[rt][ct]);
            }
        }
    }

    // ---- epilogue: +bias, BN2, ReLU, + residual x, store ----
    #pragma unroll
    for (int rt = 0; rt < 2; ++rt) {
        #pragma unroll
        for (int ct = 0; ct < 4; ++ct) {
            #pragma unroll
            for (int i = 0; i < 8; ++i) {
                const int o   = orow0 + rt * 16 + lhi * 8 + i;
                const int col = col0 + ct * 16 + lm;
                if (col < COLS) {
                    float val = acc[rt][ct][i] + tb[o];
                    const float sc = g2[o] * rsqrtf(v2[o] + BN_EPS);
                    val = fmaxf(val * sc + (b2[o] - m2[o] * sc), 0.0f);
                    const size_t idx = ((size_t)n * COUT + o) * COLS + col;
                    out[idx] = val + x[idx];
                }
            }
        }
    }
}

// ===========================================================================
extern "C" void kernel_launch(void* const* d_in, const int* in_sizes, int n_in,
                              void* d_out, int out_size, void* d_ws, size_t ws_size,
                              hipStream_t stream) {
    const float* x      = (const float*)d_in[0];
    const float* A      = (const float*)d_in[1];
    // d_in[2] = att_A (unused by the reference S_GC path)
    const float* M      = (const float*)d_in[3];
    const float* conv_w = (const float*)d_in[4];
    const float* conv_b = (const float*)d_in[5];
    const float* g1     = (const float*)d_in[6];
    const float* b1     = (const float*)d_in[7];
    const float* m1     = (const float*)d_in[8];
    const float* v1     = (const float*)d_in[9];
    const float* tw     = (const float*)d_in[10];
    const float* tb     = (const float*)d_in[11];
    const float* g2     = (const float*)d_in[12];
    const float* b2     = (const float*)d_in[13];
    const float* m2     = (const float*)d_in[14];
    const float* v2     = (const float*)d_in[15];
    float* out = (float*)d_out;
    float* h1  = (float*)d_ws;   // (N, COUT, T, V) fp32 intermediate, ~123 MB

    dim3 blk(256);
    dim3 grid1(T_, N_);
    sgc_bn_relu_kernel<<<grid1, blk, 0, stream>>>(
        x, A, M, conv_w, conv_b, g1, b1, m1, v1, h1);

    dim3 grid2((COLS + CTILE - 1) / CTILE, N_);
    tconv_bn_relu_res_kernel<<<grid2, blk, 0, stream>>>(
        h1, tw, tb, g2, b2, m2, v2, x, out);
}